// BatchTreeEncoder_76914274336962
// MI455X (gfx1250) — compile-verified
//
#include <hip/hip_runtime.h>
#include <hip/hip_bf16.h>

#define NUM_TREES 20000
#define MAX_DEPTH 8
#define MAX_SIZE 40
#define VOCAB 30000
#define ENCODE_DIM 128

typedef __attribute__((ext_vector_type(2))) float v2f;
typedef __attribute__((ext_vector_type(8))) float v8f;

// ---------------------------------------------------------------------------
// Kernel 1: edot[v] = dot(embedding[v, :], cw[:,0])  via V_WMMA_F32_16X16X4_F32
// One wave (32 threads) per 16 vocab rows. B = cw broadcast into column N=0.
// A layout (16x4 f32): lanes 0-15: M=lane, VGPR0=K0,VGPR1=K1; lanes 16-31: K2,K3.
// B layout (4x16 f32): lanes 0-15: N=lane, VGPR0=K0,VGPR1=K1; lanes 16-31: K2,K3.
// D layout (16x16 f32): VGPR i: lanes0-15 -> M=i, lanes16-31 -> M=8+i; N=lane%16.
// B loads are branch-free: every lane loads the uniform cw pair, then selects
// to zero for N!=0 (v_cndmask) -- avoids exec-mask save/branch per iteration.
// ---------------------------------------------------------------------------
__global__ __launch_bounds__(32) void edot_wmma_kernel(
    const float* __restrict__ emb,   // [VOCAB, 128]
    const float* __restrict__ cw,    // [128]
    float* __restrict__ edot)        // [VOCAB]
{
  const int r0   = blockIdx.x * 16;
  const int lane = threadIdx.x;      // 0..31
  const int half = lane >> 4;        // 0 or 1
  const int m    = lane & 15;
  const bool n0  = (m == 0);         // lane holds column N=0 of B

  v8f c = {};
  const float* rowp = emb + (size_t)(r0 + m) * ENCODE_DIM;
  const float* cwp  = cw + 2 * half;

#pragma unroll
  for (int k0 = 0; k0 < ENCODE_DIM; k0 += 4) {
    v2f a, b;
    a.x = rowp[k0 + 2 * half + 0];
    a.y = rowp[k0 + 2 * half + 1];
    const float bx = cwp[k0 + 0];    // uniform load, always executed
    const float by = cwp[k0 + 1];
    b.x = n0 ? bx : 0.f;             // v_cndmask, no divergent branch
    b.y = n0 ? by : 0.f;
    c = __builtin_amdgcn_wmma_f32_16x16x4_f32(
        /*neg_a=*/false, a, /*neg_b=*/false, b,
        /*c_mod=*/(short)0, c, /*reuse_a=*/false, /*reuse_b=*/false);
  }

  // Column N=0 of D: lane 0 holds rows r0..r0+7 in c[0..7], lane 16 rows r0+8..r0+15
  if (n0) {
#pragma unroll
    for (int i = 0; i < 8; ++i) edot[r0 + half * 8 + i] = c[i];
  }
}

// ---------------------------------------------------------------------------
// Kernel 2: per-tree encoder. One block (128 threads = 4 waves) per tree.
// Thread t owns output dim t. Uses the algebraic identity:
//   h_s . cw        = edot[tok_s] + gate_s*child_s*(pooled . cw)
//   pooled_new[t]   = sum_s attn_s*emb[tok_s][t] + (sum_s attn_s*gate_s*child_s)*pooled[t]
// ---------------------------------------------------------------------------
__global__ __launch_bounds__(128) void tree_encode_kernel(
    const int* __restrict__ tokens,          // [N, D, S]
    const unsigned char* __restrict__ masks, // [N, D, S] (bool)
    const float* __restrict__ emb,           // [VOCAB, 128]
    const float* __restrict__ cw,            // [128]
    const float* __restrict__ edot,          // [VOCAB]
    float* __restrict__ out)                 // [N, 128]
{
  const int n    = blockIdx.x;
  const int t    = threadIdx.x;   // 0..127
  const int lane = t & 31;
  const int wave = t >> 5;

  __shared__ float s_attn[MAX_SIZE];
  __shared__ float s_logit[MAX_SIZE];
  __shared__ float s_gc[MAX_SIZE];     // gate_s * child_s (0 when not a child)
  __shared__ int   s_tok[MAX_SIZE];    // -1 when invalid
  __shared__ float s_part[4];
  __shared__ float s_c;                // broadcast of sum_s attn*gate*child
  __shared__ int   s_cnt[MAX_DEPTH];

  const float cwt  = cw[t];
  const int   base = n * MAX_DEPTH * MAX_SIZE;
  const float NEG_INF = -__builtin_inff();

  // per-level mask counts (for nc of level d+1)
  if (t < MAX_DEPTH) {
    int cnt = 0;
    for (int s = 0; s < MAX_SIZE; ++s) cnt += masks[base + t * MAX_SIZE + s] ? 1 : 0;
    s_cnt[t] = cnt;
  }

  // ---- bottom level (d = D-1): pooled = sum_s valid_s * emb[tok_s][t]
  {
    const int lbase = base + (MAX_DEPTH - 1) * MAX_SIZE;
    if (t < MAX_SIZE) {
      int tk = tokens[lbase + t];
      bool v = masks[lbase + t] && (tk >= 0);
      s_tok[t] = v ? (tk < 0 ? 0 : tk) : -1;
    }
  }
  __syncthreads();

  float pooled = 0.f;
  for (int s = 0; s < MAX_SIZE; ++s) {
    int tk = s_tok[s];
    if (tk >= 0) pooled += emb[(size_t)tk * ENCODE_DIM + t];
  }
  __syncthreads();

  // ---- levels d = D-2 .. 0
  for (int d = MAX_DEPTH - 2; d >= 0; --d) {
    const int lbase = base + d * MAX_SIZE;
    const int nc = s_cnt[d + 1] > 1 ? s_cnt[d + 1] : 1;

    if (t < MAX_SIZE) {
      int tk = tokens[lbase + t];
      bool v = masks[lbase + t] && (tk >= 0);
      int tkc = tk < 0 ? 0 : tk;
      float e = edot[tkc];
      float gate = 1.f / (1.f + __expf(-e));
      bool child = v && (t < nc);
      s_tok[t]   = v ? tkc : -1;
      s_gc[t]    = child ? gate : 0.f;
      s_logit[t] = v ? e : NEG_INF;
    }

    // p_dot = pooled . cw  (128-wide reduction: wave shuffle + 4-way shared)
    float val = pooled * cwt;
#pragma unroll
    for (int off = 16; off > 0; off >>= 1) val += __shfl_xor(val, off, 32);
    if (lane == 0) s_part[wave] = val;
    __syncthreads();
    const float p_dot = s_part[0] + s_part[1] + s_part[2] + s_part[3];

    // softmax over 40 scalars + c = sum attn*gate*child, done by wave 0
    if (wave == 0) {
      const bool has2 = (lane + 32) < MAX_SIZE;
      float l0 = s_logit[lane] + s_gc[lane] * p_dot;
      float l1 = has2 ? (s_logit[lane + 32] + s_gc[lane + 32] * p_dot) : NEG_INF;
      float mx = fmaxf(l0, l1);
#pragma unroll
      for (int off = 16; off > 0; off >>= 1) mx = fmaxf(mx, __shfl_xor(mx, off, 32));
      float e0 = __expf(l0 - mx);
      float e1 = has2 ? __expf(l1 - mx) : 0.f;
      float sum = e0 + e1;
#pragma unroll
      for (int off = 16; off > 0; off >>= 1) sum += __shfl_xor(sum, off, 32);
      const float inv = 1.f / sum;
      const float a0 = e0 * inv;
      const float a1 = e1 * inv;
      s_attn[lane] = a0;
      if (has2) s_attn[lane + 32] = a1;
      float cv = a0 * s_gc[lane] + (has2 ? a1 * s_gc[lane + 32] : 0.f);
#pragma unroll
      for (int off = 16; off > 0; off >>= 1) cv += __shfl_xor(cv, off, 32);
      if (lane == 0) s_c = cv;
    }
    __syncthreads();

    // pooled_new[t] = sum_s attn_s * emb[tok_s][t] + c * pooled[t]
    float acc = 0.f;
    for (int s = 0; s < MAX_SIZE; ++s) {
      int tk = s_tok[s];
      if (tk >= 0) acc = fmaf(s_attn[s], emb[(size_t)tk * ENCODE_DIM + t], acc);
    }
    pooled = acc + s_c * pooled;
    __syncthreads();   // shared arrays are rewritten next iteration
  }

  out[(size_t)n * ENCODE_DIM + t] = pooled;
}

extern "C" void kernel_launch(void* const* d_in, const int* in_sizes, int n_in,
                              void* d_out, int out_size, void* d_ws, size_t ws_size,
                              hipStream_t stream) {
  const int*           tokens = (const int*)d_in[0];
  const unsigned char* masks  = (const unsigned char*)d_in[1];
  const float*         emb    = (const float*)d_in[2];
  const float*         cw     = (const float*)d_in[3];
  float* out  = (float*)d_out;
  float* edot = (float*)d_ws;   // 30000 floats = 120 KB scratch

  edot_wmma_kernel<<<VOCAB / 16, 32, 0, stream>>>(emb, cw, edot);
  tree_encode_kernel<<<NUM_TREES, 128, 0, stream>>>(tokens, masks, emb, cw, edot, out);
}